// TrilinearLUTFunction_2448131359063
// MI455X (gfx1250) — compile-verified
//
#include <hip/hip_runtime.h>

// Trilinear 3D-LUT color transform, MI455X (gfx1250, wave32).
// img: (B, 3, H, W) f32 planar, lut: (B, 33, 33, 33, 3) f32, out: (B, 3, H, W) f32.
// Memory-bound: ~201 MB stream -> ~8.6 us floor at 23.3 TB/s; LUT lives in L2.

typedef float f4 __attribute__((ext_vector_type(4)));

#define LUT_G      33
#define LUT_ENTRY  3
#define LUT_FLOATS (LUT_G * LUT_G * LUT_G * LUT_ENTRY)   // 107811 floats / batch
#define CLIP_HI    (32.0f - 1e-6f)                        // g-1-1e-6

__device__ __forceinline__ void load6(const float* __restrict__ p,
                                      float c0[3], float c1[3]) {
  // 6 contiguous floats: LUT entries (x,y,z0) and (x,y,z0+1).
  c0[0] = p[0]; c0[1] = p[1]; c0[2] = p[2];
  c1[0] = p[3]; c1[1] = p[4]; c1[2] = p[5];
}

__global__ __launch_bounds__(256) void
trilinear_lut_kernel(const float* __restrict__ img,
                     const float* __restrict__ lut,
                     float* __restrict__ out,
                     int HW) {
  const int b   = blockIdx.y;
  const int HW4 = HW >> 2;
  const int v   = blockIdx.x * 256 + threadIdx.x;   // 4-pixel vector index

  const float* __restrict__ lutb = lut + (size_t)b * LUT_FLOATS;

  // --- Warm the per-batch LUT into L2 (global_prefetch_b8, gfx1250). ---
  // 431,244 bytes / batch; first ~3369 lanes of the grid each prefetch one
  // 128-B line. Speculative, no counter cost, keeps the gather storm in L2.
  {
    const size_t lutBytes = (size_t)LUT_FLOATS * sizeof(float);
    const size_t poff     = (size_t)v * 128;
    if (poff < lutBytes)
      __builtin_prefetch((const char*)lutb + poff, 0, 3);
  }

  if (v >= HW4) return;
  const size_t px = (size_t)v * 4;

  // --- Streaming planar loads, non-temporal (TH=NT) so the LUT stays hot. ---
  const float* __restrict__ imgb = img + (size_t)b * 3 * (size_t)HW;
  const f4 r4 = __builtin_nontemporal_load((const f4*)(imgb            + px));
  const f4 g4 = __builtin_nontemporal_load((const f4*)(imgb +     HW   + px));
  const f4 b4 = __builtin_nontemporal_load((const f4*)(imgb + 2 * (size_t)HW + px));

  f4 oR, oG, oB;

#pragma unroll
  for (int j = 0; j < 4; ++j) {
    const float x = fminf(fmaxf(r4[j] * 32.0f, 0.0f), CLIP_HI);
    const float y = fminf(fmaxf(g4[j] * 32.0f, 0.0f), CLIP_HI);
    const float z = fminf(fmaxf(b4[j] * 32.0f, 0.0f), CLIP_HI);

    const int x0 = (int)x, y0 = (int)y, z0 = (int)z;   // floor (x,y,z >= 0)
    const float xd = x - (float)x0;
    const float yd = y - (float)y0;
    const float zd = z - (float)z0;
    const float xi = 1.0f - xd, yi = 1.0f - yd, zi = 1.0f - zd;

    // After the clip, x0 <= 31 so x1 = x0+1 (no clamp needed); z-pairs are
    // contiguous 24-B runs -> 4 gathers of 6 floats per pixel.
    const int i00 = ((x0 * LUT_G + y0) * LUT_G + z0) * LUT_ENTRY;
    const int i10 = i00 + LUT_G * LUT_G * LUT_ENTRY;   // x+1
    const int i01 = i00 + LUT_G * LUT_ENTRY;           // y+1
    const int i11 = i10 + LUT_G * LUT_ENTRY;           // x+1, y+1

    float c000[3], c001[3], c100[3], c101[3];
    float c010[3], c011[3], c110[3], c111[3];
    load6(lutb + i00, c000, c001);
    load6(lutb + i10, c100, c101);
    load6(lutb + i01, c010, c011);
    load6(lutb + i11, c110, c111);

    float o[3];
#pragma unroll
    for (int c = 0; c < 3; ++c) {
      const float c00 = c000[c] * xi + c100[c] * xd;
      const float c01 = c001[c] * xi + c101[c] * xd;
      const float c10 = c010[c] * xi + c110[c] * xd;
      const float c11 = c011[c] * xi + c111[c] * xd;
      const float c0  = c00 * yi + c10 * yd;
      const float c1  = c01 * yi + c11 * yd;
      o[c] = c0 * zi + c1 * zd;
    }
    oR[j] = o[0];
    oG[j] = o[1];
    oB[j] = o[2];
  }

  // --- Streaming planar stores, non-temporal. ---
  float* __restrict__ outb = out + (size_t)b * 3 * (size_t)HW;
  __builtin_nontemporal_store(oR, (f4*)(outb                + px));
  __builtin_nontemporal_store(oG, (f4*)(outb +     HW       + px));
  __builtin_nontemporal_store(oB, (f4*)(outb + 2 * (size_t)HW + px));
}

extern "C" void kernel_launch(void* const* d_in, const int* in_sizes, int n_in,
                              void* d_out, int out_size, void* d_ws, size_t ws_size,
                              hipStream_t stream) {
  const float* img = (const float*)d_in[0];
  const float* lut = (const float*)d_in[1];
  float*       out = (float*)d_out;

  // Derive shapes: lut is (B, 33,33,33,3); img is (B, 3, H, W).
  const int B  = in_sizes[1] / LUT_FLOATS;          // = 8
  const int HW = in_sizes[0] / (3 * B);             // = 1024*1024

  const int HW4    = HW >> 2;                        // 4 pixels per thread
  const int blocks = (HW4 + 255) / 256;              // 1024 blocks per image

  dim3 grid(blocks, B);
  trilinear_lut_kernel<<<grid, 256, 0, stream>>>(img, lut, out, HW);
}